// AttentionUnitLayer_33440615367298
// MI455X (gfx1250) — compile-verified
//
#include <hip/hip_runtime.h>
#include <math.h>

// Problem constants (match reference): B=4096, T=204800, D=32, H=64
// in_dim = 1088; Q tile per candidate: [D=32 (i), H=64 (k)] -> N = D*H = 2048
#define DD 32
#define HH 64
#define NQ (DD * HH)   // 2048
#define EPSF 1e-10f

typedef __attribute__((ext_vector_type(2))) float v2f;
typedef __attribute__((ext_vector_type(8))) float v8f;

// D = A(16x4 f32) * B(4x16 f32) + C(16x16 f32), exact fp32 WMMA on CDNA5
__device__ __forceinline__ v8f wmma4(v2f a, v2f b, v8f c) {
  return __builtin_amdgcn_wmma_f32_16x16x4_f32(
      /*neg_a=*/false, a, /*neg_b=*/false, b,
      /*c_mod=*/(short)0, c, /*reuse_a=*/false, /*reuse_b=*/false);
}

// ---------------------------------------------------------------------------
// Kernel 1: Q[b, i*64+k] = W1_b[i,k] + sum_j cand[b,j] * G[i,j,k]
//   GEMM: M=4096 (b), K=32 (j), N=2048 (i*64+k)
//   G[i,j,k] = W1[(64 + i*32 + j)*64 + k]  (flat index 4096 + i*2048 + j*64 + k)
//   One wave computes one 16x16 tile via 8x V_WMMA_F32_16X16X4_F32.
// ---------------------------------------------------------------------------
__global__ __launch_bounds__(32) void k1_build_q(
    const float* __restrict__ cand, const float* __restrict__ W1,
    float* __restrict__ Q) {
  const int lane = threadIdx.x;
  const int lo = lane & 15;
  const int hi = lane >> 4;           // 0 or 1
  const int n0 = blockIdx.x * 16;     // N tile (0..2047)
  const int m0 = blockIdx.y * 16;     // M tile (0..4095)
  const int n = n0 + lo;              // this lane's output column
  const int i = n >> 6;               // i = n / 64
  const int k = n & 63;               // k = n % 64
  const int mrow = m0 + lo;           // this lane's A row

  v8f acc = {};
#pragma unroll
  for (int c = 0; c < 8; ++c) {       // K chunks of 4 (j = c*4 .. c*4+3)
    const int j0 = c * 4 + hi * 2;
    v2f a;
    a.x = cand[mrow * DD + j0];
    a.y = cand[mrow * DD + j0 + 1];
    v2f b;
    b.x = W1[4096 + i * 2048 + (j0)     * HH + k];
    b.y = W1[4096 + i * 2048 + (j0 + 1) * HH + k];
    acc = wmma4(a, b, acc);
  }
  // fold in W1_b bias (depends only on column n -> same for all 8 rows)
  const float bias = W1[(DD + i) * HH + k];
#pragma unroll
  for (int r = 0; r < 8; ++r) {
    Q[(size_t)(m0 + r + 8 * hi) * NQ + n] = acc[r] + bias;
  }
}

// ---------------------------------------------------------------------------
// Kernel 2: one 32-thread block per candidate b (segment of sorted row_ids).
//   H[t,k] = c1_b[k] + sum_i beh[t,i] * Q_b[i,k]   (WMMA, M-tiles of 16 rows)
//   dice + h@W2 + b2 -> w_t ; out[b,:] = sum_t w_t * beh[t,:]
// ---------------------------------------------------------------------------
__global__ __launch_bounds__(32) void k2_segments(
    const float* __restrict__ cand, const float* __restrict__ beh,
    const int* __restrict__ rid, const float* __restrict__ W1,
    const float* __restrict__ b1, const float* __restrict__ alpha,
    const float* __restrict__ W2, const float* __restrict__ b2,
    const float* __restrict__ Q, float* __restrict__ out, int T) {
  __shared__ float Hld[16][HH + 1];   // padded stride: bank-conflict free
  __shared__ float wld[16];

  const int b = blockIdx.x;
  const int lane = threadIdx.x;
  const int lo = lane & 15;
  const int hi = lane >> 4;

  // binary search segment [start, end) of row_ids == b (row_ids sorted)
  int a0 = 0, a1 = T;
  while (a0 < a1) { int mid = (a0 + a1) >> 1; if (rid[mid] < b) a0 = mid + 1; else a1 = mid; }
  const int start = a0;
  a1 = T;
  while (a0 < a1) { int mid = (a0 + a1) >> 1; if (rid[mid] <= b) a0 = mid + 1; else a1 = mid; }
  const int end = a0;

  // Preload B fragments of Q_b: K-dim = i (32, 8 chunks of 4), N-dim = k (4 tiles of 16)
  float bq[4][8][2];
#pragma unroll
  for (int nt = 0; nt < 4; ++nt)
#pragma unroll
    for (int c = 0; c < 8; ++c) {
      const int j0 = c * 4 + hi * 2;
      bq[nt][c][0] = Q[(size_t)b * NQ + (j0)     * HH + nt * 16 + lo];
      bq[nt][c][1] = Q[(size_t)b * NQ + (j0 + 1) * HH + nt * 16 + lo];
    }

  // c1_b[k] = b1[k] + cand_b . W1_c[:,k]  for this lane's 4 columns
  float c1v[4];
#pragma unroll
  for (int nt = 0; nt < 4; ++nt) c1v[nt] = b1[nt * 16 + lo];
  for (int j = 0; j < DD; ++j) {
    const float cj = cand[b * DD + j];
#pragma unroll
    for (int nt = 0; nt < 4; ++nt) c1v[nt] += cj * W1[j * HH + nt * 16 + lo];
  }

  float acc = 0.0f;                   // out[b, d=lane] accumulator
  const int rr = lane >> 1;           // dice: 2 lanes per row
  const int kh = (lane & 1) * 32;     // each lane covers 32 of 64 columns

  for (int t0 = start; t0 < end; t0 += 16) {
    // accumulators seeded with c1 (all 8 rows of a tile share the same k)
    v8f at[4];
#pragma unroll
    for (int nt = 0; nt < 4; ++nt) {
      v8f t;
#pragma unroll
      for (int r = 0; r < 8; ++r) t[r] = c1v[nt];
      at[nt] = t;
    }
    // GEMM: 8 K-chunks x 4 N-tiles
#pragma unroll
    for (int c = 0; c < 8; ++c) {
      const int row = t0 + lo;
      const int j0 = c * 4 + hi * 2;
      v2f a;
      if (row < end) {
        a.x = beh[(size_t)row * DD + j0];
        a.y = beh[(size_t)row * DD + j0 + 1];
      } else {
        a.x = 0.0f; a.y = 0.0f;
      }
#pragma unroll
      for (int nt = 0; nt < 4; ++nt) {
        v2f bb; bb.x = bq[nt][c][0]; bb.y = bq[nt][c][1];
        at[nt] = wmma4(a, bb, at[nt]);
      }
    }
    // spill H tile to LDS: element (r,L) holds row t0 + r + 8*hi, col nt*16+lo
#pragma unroll
    for (int nt = 0; nt < 4; ++nt)
#pragma unroll
      for (int r = 0; r < 8; ++r)
        Hld[r + 8 * hi][nt * 16 + lo] = at[nt][r];
    __syncthreads();

    // Dice: mean/std over 64 cols (pair of lanes per row, shfl_xor combine)
    float s1 = 0.0f, s2 = 0.0f;
    for (int j = 0; j < 32; ++j) {
      const float x = Hld[rr][kh + j];
      s1 += x; s2 += x * x;
    }
    s1 += __shfl_xor(s1, 1);
    s2 += __shfl_xor(s2, 1);
    const float m   = s1 * (1.0f / 64.0f);
    const float var = s2 * (1.0f / 64.0f) - m * m;
    const float sd  = sqrtf(var + EPSF);       // mean((x-m)^2 + EPS)
    const float inv = 1.0f / (sd + EPSF);

    float wacc = 0.0f;
    for (int j = 0; j < 32; ++j) {
      const float x = Hld[rr][kh + j];
      const float z = (x - m) * inv;
      const float p = 1.0f / (1.0f + __expf(-z));
      const float d = alpha[kh + j] * (1.0f - p) * x + p * x;
      wacc += d * W2[kh + j];
    }
    wacc += __shfl_xor(wacc, 1);
    float wv = wacc + b2[0];
    if (t0 + rr >= end) wv = 0.0f;             // mask padded rows
    wld[rr] = wv;                              // both lanes write same value
    __syncthreads();

    // weighted segment-sum: lane = output dim d
    const int nrows = (end - t0) < 16 ? (end - t0) : 16;
    for (int r2 = 0; r2 < nrows; ++r2) {
      acc += wld[r2] * beh[(size_t)(t0 + r2) * DD + lane];
    }
    __syncthreads();                           // protect Hld/wld reuse
  }

  out[b * DD + lane] = acc;                    // block exclusively owns row b
}

extern "C" void kernel_launch(void* const* d_in, const int* in_sizes, int n_in,
                              void* d_out, int out_size, void* d_ws, size_t ws_size,
                              hipStream_t stream) {
  const float* cand  = (const float*)d_in[0];
  const float* beh   = (const float*)d_in[1];
  const int*   rid   = (const int*)d_in[2];
  const float* W1    = (const float*)d_in[3];
  const float* b1    = (const float*)d_in[4];
  const float* alpha = (const float*)d_in[5];
  const float* W2    = (const float*)d_in[6];
  const float* b2    = (const float*)d_in[7];
  float* outp = (float*)d_out;
  float* Q = (float*)d_ws;                     // B * 2048 floats = 33.5 MB

  const int Bn = in_sizes[0] / DD;             // 4096
  const int T  = in_sizes[2];                  // 204800

  dim3 g1(NQ / 16, Bn / 16);                   // 128 x 256 tiles
  k1_build_q<<<g1, 32, 0, stream>>>(cand, W1, Q);
  k2_segments<<<Bn, 32, 0, stream>>>(cand, beh, rid, W1, b1, alpha, W2, b2, Q,
                                     outp, T);
}